// PeriodSparseSelfAttention_45140106281299
// MI455X (gfx1250) — compile-verified
//
#include <hip/hip_runtime.h>
#include <math.h>
#include <stdint.h>

typedef __attribute__((ext_vector_type(2))) float v2f;
typedef __attribute__((ext_vector_type(8))) float v8f;
typedef __attribute__((ext_vector_type(4))) unsigned int v4u;
typedef __attribute__((ext_vector_type(4))) int v4i;
typedef __attribute__((ext_vector_type(8))) int v8i;

// Problem constants (match reference)
constexpr int Bn = 2, Tn = 2048, Dn = 1024, Hn = 16, HDn = 64;

// GEMM tiling
constexpr int TM = 128, TN = 64, TK = 16;
constexpr int ASTR = 17;  // A tile LDS row stride (16 + 1 pad dword via TDM)
constexpr int BSTR = 68;  // 64-wide tile LDS row stride (64 + 4 pad dwords via TDM)

__device__ __forceinline__ v8f wmma_f32(v2f a, v2f b, v8f c) {
  // D = A(16x4 f32) * B(4x16 f32) + C(16x16 f32)
  return __builtin_amdgcn_wmma_f32_16x16x4_f32(
      false, a, false, b, (short)0, c, false, false);
}

// Flat addresses of __shared__ objects carry the LDS offset in the low 32 bits.
__device__ __forceinline__ uint32_t lds_addr_of(const void* p) {
  return (uint32_t)(uintptr_t)p;
}

// ---------------------------------------------------------------------------
// Tensor Data Mover: load a (rows x dim0) f32 tile (global row stride
// rowStride elements) into LDS at ldsByte, inserting padAmount(+1) dwords of
// padding after every 2^(padInterval+1) dwords (ISA 8.3/8.4 D# encoding).
// Wave-level DMA; tracked by TENSORcnt.
// ---------------------------------------------------------------------------
__device__ __forceinline__ void tdm_load_2d(const float* gsrc, uint32_t ldsByte,
                                            uint32_t dim0, uint32_t rows,
                                            uint64_t rowStride,
                                            uint32_t padInterval,
                                            uint32_t padAmount) {
  const uint64_t ga = (uint64_t)(uintptr_t)gsrc;
  const uint64_t d1s = rowStride * rows;  // dim1 stride (unused for 2D tiles)
  v4u g0;
  g0.x = 1u;                                 // count=1: valid user descriptor
  g0.y = ldsByte;                            // lds_addr
  g0.z = (uint32_t)ga;                       // global_addr[31:0]
  g0.w = (uint32_t)(ga >> 32) | (2u << 30);  // global_addr[56:32] | type=2
  v8i g1;
  g1[0] = (int)((2u << 16) | (1u << 20) | (padInterval << 22) | (padAmount << 25));
  g1[1] = (int)(dim0 << 16);                              // tensor_dim0[15:0]
  g1[2] = (int)((dim0 >> 16) | (rows << 16));             // tdim0 hi | tdim1 lo
  g1[3] = (int)((rows >> 16) | (dim0 << 16));             // tdim1 hi | tile_dim0
  g1[4] = (int)(rows & 0xFFFFu);                          // tile_dim1 (tile_dim2=0)
  g1[5] = (int)(uint32_t)rowStride;                       // tdim0_stride[31:0]
  g1[6] = (int)(((uint32_t)(rowStride >> 32) & 0xFFFFu) | ((uint32_t)d1s << 16));
  g1[7] = (int)(uint32_t)(d1s >> 16);                     // tdim1_stride[47:16]
  v4i gz = {0, 0, 0, 0};
#if __clang_major__ >= 23
  v8i gz8 = {0, 0, 0, 0, 0, 0, 0, 0};
  __builtin_amdgcn_tensor_load_to_lds(g0, g1, gz, gz, gz8, 0);
#else
  __builtin_amdgcn_tensor_load_to_lds(g0, g1, gz, gz, 0);
#endif
}

// ---------------------------------------------------------------------------
// fp32 WMMA GEMM, TDM double-buffered: Out = A(MxK) * B(KxN) + bias
// MODE 0: QKV epilogue — scatter to (3,B,H,T,HD), Q pre-scaled by 1/sqrt(HD)
// MODE 1: plain row-major MxN output
// ---------------------------------------------------------------------------
template <int MODE>
__global__ __launch_bounds__(256) void gemm_f32_wmma(
    const float* __restrict__ A, const float* __restrict__ Bm,
    const float* __restrict__ bias, float* __restrict__ Out,
    int M, int N, int K) {
  __shared__ float As[2][TM][ASTR];  // m-major: As[buf][m][k]
  __shared__ float Bs[2][TK][BSTR];  // Bs[buf][k][n]

  const int tid  = threadIdx.x;
  const int lane = tid & 31;
  const int wave = tid >> 5;
  const int lh   = lane >> 4;
  const int ll   = lane & 15;
  const bool leader = (wave == 0);

  const int mBase = blockIdx.y * TM;
  const int nBase = blockIdx.x * TN;
  const int wm = (wave & 3) * 32;
  const int wn = (wave >> 2) * 32;

  if (leader) {  // prime buffer 0
    tdm_load_2d(&A[(size_t)mBase * K], lds_addr_of(&As[0][0][0]), TK, TM, K, 3, 0);
    tdm_load_2d(&Bm[nBase], lds_addr_of(&Bs[0][0][0]), TN, TK, N, 5, 3);
  }

  v8f acc[2][2];
#pragma unroll
  for (int i = 0; i < 2; ++i)
#pragma unroll
    for (int j = 0; j < 2; ++j) acc[i][j] = {};

  const int nk = K / TK;
  for (int it = 0; it < nk; ++it) {
    const int cur = it & 1;
    if (leader) __builtin_amdgcn_s_wait_tensorcnt(0);
    __syncthreads();  // publish buffer `cur`; all waves done reading `cur^1`
    if (leader && it + 1 < nk) {  // DMA next tiles while we compute
      const int kb2 = (it + 1) * TK;
      tdm_load_2d(&A[(size_t)mBase * K + kb2], lds_addr_of(&As[cur ^ 1][0][0]),
                  TK, TM, K, 3, 0);
      tdm_load_2d(&Bm[(size_t)kb2 * N + nBase], lds_addr_of(&Bs[cur ^ 1][0][0]),
                  TN, TK, N, 5, 3);
    }
#pragma unroll
    for (int k4 = 0; k4 < TK / 4; ++k4) {
      const int kl = 4 * k4 + 2 * lh;
      v2f aF[2], bF[2];
#pragma unroll
      for (int i = 0; i < 2; ++i) {
        aF[i].x = As[cur][wm + 16 * i + ll][kl];
        aF[i].y = As[cur][wm + 16 * i + ll][kl + 1];
      }
#pragma unroll
      for (int j = 0; j < 2; ++j) {
        bF[j].x = Bs[cur][kl][wn + 16 * j + ll];
        bF[j].y = Bs[cur][kl + 1][wn + 16 * j + ll];
      }
#pragma unroll
      for (int i = 0; i < 2; ++i)
#pragma unroll
        for (int j = 0; j < 2; ++j) acc[i][j] = wmma_f32(aF[i], bF[j], acc[i][j]);
    }
  }

  // Epilogue. C layout: VGPR r, lane l -> M = r + 8*(l>>4), N = l&15.
#pragma unroll
  for (int i = 0; i < 2; ++i) {
#pragma unroll
    for (int j = 0; j < 2; ++j) {
      const int col = nBase + wn + 16 * j + ll;
      const float bv = bias[col];
#pragma unroll
      for (int r = 0; r < 8; ++r) {
        const int row = mBase + wm + 16 * i + r + 8 * lh;
        float v = acc[i][j][r] + bv;
        if (MODE == 0) {
          const int which = col >> 10;  // 0=q 1=k 2=v
          const int hi = col & 1023;
          const int h = hi >> 6, hd = hi & 63;
          const int b = row / Tn, t = row % Tn;
          if (which == 0) v *= 0.125f;  // 1/sqrt(HD)
          Out[((((size_t)which * Bn + b) * Hn + h) * Tn + t) * HDn + hd] = v;
        } else {
          Out[(size_t)row * N + col] = v;
        }
      }
    }
  }
}

// ---------------------------------------------------------------------------
// Flash attention, periodic+causal mask; K/V tiles DMA'd by the TDM with
// double buffering. One block = 64 queries of one (b,h); 4 waves x 16 rows.
// ---------------------------------------------------------------------------
__global__ __launch_bounds__(128) void attn_flash_wmma(
    const float* __restrict__ Q, const float* __restrict__ Kt,
    const float* __restrict__ V, const int* __restrict__ periods,
    float* __restrict__ Out) {
  __shared__ float Ks[2][64][BSTR];
  __shared__ float Vs[2][64][BSTR];
  __shared__ float Ps[4][16][BSTR];  // per-wave C->A layout bounce buffer

  const int tid  = threadIdx.x;
  const int lane = tid & 31;
  const int wave = tid >> 5;
  const int lh   = lane >> 4;
  const int ll   = lane & 15;
  const bool leader = (wave == 0);

  const int bh = blockIdx.y;
  const int b  = bh >> 4;
  const int h  = bh & 15;
  const int qt = blockIdx.x;
  const int qbase = qt * 64;
  const int m0 = wave * 16;

  int p = periods[bh];
  if (p < 1) p = 1;

  const size_t headOff = (size_t)bh * Tn * HDn;

  if (leader) {  // prime buffer 0 with key/value tile 0 (contiguous 16KB each)
    tdm_load_2d(Kt + headOff, lds_addr_of(&Ks[0][0][0]), 64, 64, 64, 5, 3);
    tdm_load_2d(V + headOff, lds_addr_of(&Vs[0][0][0]), 64, 64, 64, 5, 3);
  }

  // Preload Q fragments (loop-invariant A-matrix): 16 k-steps over HD=64
  v2f aQ[16];
  {
    const float* qrow = Q + headOff + (size_t)(qbase + m0 + ll) * HDn;
#pragma unroll
    for (int kk = 0; kk < 16; ++kk) {
      aQ[kk].x = qrow[4 * kk + 2 * lh];
      aQ[kk].y = qrow[4 * kk + 2 * lh + 1];
    }
  }

  v8f oacc[4];
#pragma unroll
  for (int j = 0; j < 4; ++j) oacc[j] = {};
  float rmax[8], rsum[8];
#pragma unroll
  for (int r = 0; r < 8; ++r) { rmax[r] = -INFINITY; rsum[r] = 0.0f; }

  for (int jt = 0; jt <= qt; ++jt) {
    const int cur = jt & 1;
    const int kb = jt * 64;
    if (leader) __builtin_amdgcn_s_wait_tensorcnt(0);
    __syncthreads();  // publish buffer `cur`
    if (leader && jt < qt) {  // DMA next K/V tile during compute
      const size_t off = headOff + (size_t)(kb + 64) * HDn;
      tdm_load_2d(Kt + off, lds_addr_of(&Ks[cur ^ 1][0][0]), 64, 64, 64, 5, 3);
      tdm_load_2d(V + off, lds_addr_of(&Vs[cur ^ 1][0][0]), 64, 64, 64, 5, 3);
    }

    // S = Q @ K^T : 16 x 64 per wave (Q pre-scaled by 1/sqrt(HD))
    v8f s[4];
#pragma unroll
    for (int j = 0; j < 4; ++j) s[j] = {};
#pragma unroll
    for (int kk = 0; kk < 16; ++kk) {
      const int kl = 4 * kk + 2 * lh;
#pragma unroll
      for (int j = 0; j < 4; ++j) {
        v2f bK;
        bK.x = Ks[cur][16 * j + ll][kl];  // K^T B-fragment == row-major K read
        bK.y = Ks[cur][16 * j + ll][kl + 1];
        s[j] = wmma_f32(aQ[kk], bK, s[j]);
      }
    }

    // Periodic + causal mask
#pragma unroll
    for (int j = 0; j < 4; ++j) {
      const int jg = kb + 16 * j + ll;
#pragma unroll
      for (int r = 0; r < 8; ++r) {
        const int ig = qbase + m0 + r + 8 * lh;
        const bool valid = (jg <= ig) && (((ig - jg) % p) == 0);
        if (!valid) s[j][r] = -INFINITY;
      }
    }

    // Online softmax: per-row max/sum via 16-lane xor-shuffle reductions
#pragma unroll
    for (int r = 0; r < 8; ++r) {
      float m = rmax[r];
#pragma unroll
      for (int j = 0; j < 4; ++j) m = fmaxf(m, s[j][r]);
#pragma unroll
      for (int off = 1; off < 16; off <<= 1)
        m = fmaxf(m, __shfl_xor(m, off, 32));
      const float sc = __expf(rmax[r] - m);
      rmax[r] = m;

      float part = 0.0f;
#pragma unroll
      for (int j = 0; j < 4; ++j) {
        const float pv = __expf(s[j][r] - m);  // masked -> exp(-inf) = 0
        Ps[wave][r + 8 * lh][16 * j + ll] = pv;
        part += pv;
      }
#pragma unroll
      for (int off = 1; off < 16; off <<= 1)
        part += __shfl_xor(part, off, 32);
      rsum[r] = rsum[r] * sc + part;
#pragma unroll
      for (int j = 0; j < 4; ++j) oacc[j][r] *= sc;
    }

    // O += P @ V (same-wave LDS write->read is in-order; no barrier needed)
#pragma unroll
    for (int kk = 0; kk < 16; ++kk) {
      const int kl = 4 * kk + 2 * lh;
      v2f aP;
      aP.x = Ps[wave][ll][kl];
      aP.y = Ps[wave][ll][kl + 1];
#pragma unroll
      for (int j = 0; j < 4; ++j) {
        v2f bV;
        bV.x = Vs[cur][kl][16 * j + ll];
        bV.y = Vs[cur][kl + 1][16 * j + ll];
        oacc[j] = wmma_f32(aP, bV, oacc[j]);
      }
    }
  }

  // Normalize; write attn output in (B, T, H*HD) layout for the out-proj GEMM
#pragma unroll
  for (int r = 0; r < 8; ++r) {
    const float inv = 1.0f / rsum[r];
    const int t = qbase + m0 + r + 8 * lh;
#pragma unroll
    for (int j = 0; j < 4; ++j) {
      const int d = h * HDn + 16 * j + ll;
      Out[((size_t)b * Tn + t) * Dn + d] = oacc[j][r] * inv;
    }
  }
}

extern "C" void kernel_launch(void* const* d_in, const int* in_sizes, int n_in,
                              void* d_out, int out_size, void* d_ws, size_t ws_size,
                              hipStream_t stream) {
  (void)in_sizes; (void)n_in; (void)out_size; (void)ws_size;
  const float* x     = (const float*)d_in[0];
  const int*   per   = (const int*)d_in[1];
  const float* w_qkv = (const float*)d_in[2];
  const float* b_qkv = (const float*)d_in[3];
  const float* w_out = (const float*)d_in[4];
  const float* b_out = (const float*)d_in[5];
  float* out = (float*)d_out;

  // Workspace: [Q|K|V] in (B,H,T,HD) + attn-out in (B,T,D). 64 MB total.
  float* qkv = (float*)d_ws;
  const size_t headElems = (size_t)Bn * Hn * Tn * HDn;
  float* attn = qkv + 3 * headElems;

  dim3 g1((3 * Dn) / TN, (Bn * Tn) / TM);
  gemm_f32_wmma<0><<<g1, 256, 0, stream>>>(x, w_qkv, b_qkv, qkv,
                                           Bn * Tn, 3 * Dn, Dn);

  dim3 g2(Tn / 64, Bn * Hn);
  attn_flash_wmma<<<g2, 128, 0, stream>>>(qkv, qkv + headElems,
                                          qkv + 2 * headElems, per, attn);

  dim3 g3(Dn / TN, (Bn * Tn) / TM);
  gemm_f32_wmma<1><<<g3, 256, 0, stream>>>(attn, w_out, b_out, out,
                                           Bn * Tn, Dn, Dn);
}